// LabelLoss_5738076307626
// MI455X (gfx1250) — compile-verified
//
#include <hip/hip_runtime.h>
#include <math.h>

// Problem constants (from reference): B=32, J=64, C=7, H=W=128
#define NB   32
#define NJ   64
#define NC   7
#define NHW  16384   // 128*128
#define SMOOTH_K (0.1f / 7.0f)

typedef float v2f __attribute__((ext_vector_type(2)));
typedef float v8f __attribute__((ext_vector_type(8)));

// ---------------------------------------------------------------------------
// Kernel 1: one workgroup per (b,j). Stream 64KB of heatmap with b128 loads,
// find argmax (first-index tie break), then fuse the 7-class gather + smoothed
// BCE and emit per_obj[b*64+j].
// ---------------------------------------------------------------------------
__global__ __launch_bounds__(256) void argmax_bce_kernel(
    const float* __restrict__ pred,     // (B, C, H, W)
    const float* __restrict__ gt,       // (B, J, C)
    const float* __restrict__ heat,     // (B, J, H, W)
    float* __restrict__ per_obj)        // (B*J)
{
    const int bj = blockIdx.x;          // b*NJ + j
    const int b  = bj >> 6;
    const int t  = threadIdx.x;

    const float* hp = heat + (size_t)bj * NHW;

    float bv = -INFINITY;
    int   bi = 0x7fffffff;

    // 16 chunks * 256 threads * float4 = 16384 elements, fully coalesced.
    #pragma unroll
    for (int c = 0; c < 16; ++c) {
        const int base = (c * 256 + t) * 4;
        const float4 v = *reinterpret_cast<const float4*>(hp + base);
        if (v.x > bv || (v.x == bv && base     < bi)) { bv = v.x; bi = base;     }
        if (v.y > bv || (v.y == bv && base + 1 < bi)) { bv = v.y; bi = base + 1; }
        if (v.z > bv || (v.z == bv && base + 2 < bi)) { bv = v.z; bi = base + 2; }
        if (v.w > bv || (v.w == bv && base + 3 < bi)) { bv = v.w; bi = base + 3; }
    }

    // wave32 butterfly reduction (lane shuffles -> ds_permute path)
    #pragma unroll
    for (int off = 16; off > 0; off >>= 1) {
        const float ov = __shfl_xor(bv, off, 32);
        const int   oi = __shfl_xor(bi, off, 32);
        if (ov > bv || (ov == bv && oi < bi)) { bv = ov; bi = oi; }
    }

    __shared__ float sv[8];
    __shared__ int   si[8];
    const int wave = t >> 5;
    if ((t & 31) == 0) { sv[wave] = bv; si[wave] = bi; }
    __syncthreads();

    if (t == 0) {
        #pragma unroll
        for (int w = 1; w < 8; ++w) {
            if (sv[w] > bv || (sv[w] == bv && si[w] < bi)) { bv = sv[w]; bi = si[w]; }
        }
        // flat idx bi -> pred[b, c, h, w] == pred[b*C*HW + c*HW + bi]
        const float* pp = pred + (size_t)b * NC * NHW + bi;
        const float* gp = gt   + (size_t)bj * NC;
        float s = 0.0f;
        #pragma unroll
        for (int c = 0; c < NC; ++c) {
            const float z  = pp[c * NHW];
            const float td = 0.9f * gp[c] + SMOOTH_K;
            s += fmaxf(z, 0.0f) - z * td + log1pf(expf(-fabsf(z)));
        }
        per_obj[bj] = s;
    }
}

// ---------------------------------------------------------------------------
// Kernel 2: out(32) = per_obj(32x64) x ones(64)/64 via V_WMMA_F32_16X16X4_F32.
// One wave. Two M-tiles (rows 0..15, 16..31), 16 chained K-blocks of 4.
// B is the constant 1/64 in every slot, so every column of D equals the row
// mean; we read column N=0 per the documented C/D layout.
// ---------------------------------------------------------------------------
__global__ __launch_bounds__(32) void mean_wmma_kernel(
    const float* __restrict__ per_obj,  // (32, 64)
    float* __restrict__ out)            // (32)
{
    const int l    = threadIdx.x;
    const int row  = l & 15;            // A: lanes 0-15 and 16-31 both hold M=0..15
    const int koff = (l >> 4) * 2;      // lanes 16-31 hold K=+2,+3 of the 4-block

    const v2f bvec = { 1.0f / 64.0f, 1.0f / 64.0f };
    v8f c0 = {0.f, 0.f, 0.f, 0.f, 0.f, 0.f, 0.f, 0.f};
    v8f c1 = {0.f, 0.f, 0.f, 0.f, 0.f, 0.f, 0.f, 0.f};

    #pragma unroll
    for (int kb = 0; kb < 16; ++kb) {
        const float* p0 = per_obj + row * 64        + kb * 4 + koff;  // M-tile 0
        const float* p1 = per_obj + (row + 16) * 64 + kb * 4 + koff;  // M-tile 1
        const v2f a0 = { p0[0], p0[1] };
        const v2f a1 = { p1[0], p1[1] };
        c0 = __builtin_amdgcn_wmma_f32_16x16x4_f32(false, a0, false, bvec,
                                                   (short)0, c0, false, false);
        c1 = __builtin_amdgcn_wmma_f32_16x16x4_f32(false, a1, false, bvec,
                                                   (short)0, c1, false, false);
    }

    // C/D layout: VGPR r, lanes 0-15 -> M=r ; lanes 16-31 -> M=r+8 (N = lane%16)
    if (l == 0) {
        #pragma unroll
        for (int r = 0; r < 8; ++r) { out[r]      = c0[r]; out[16 + r] = c1[r]; }
    } else if (l == 16) {
        #pragma unroll
        for (int r = 0; r < 8; ++r) { out[8 + r]  = c0[r]; out[24 + r] = c1[r]; }
    }
}

// ---------------------------------------------------------------------------
extern "C" void kernel_launch(void* const* d_in, const int* in_sizes, int n_in,
                              void* d_out, int out_size, void* d_ws, size_t ws_size,
                              hipStream_t stream) {
    (void)in_sizes; (void)n_in; (void)out_size; (void)ws_size;
    const float* pred = (const float*)d_in[0];   // (32, 7, 128, 128)
    const float* gt   = (const float*)d_in[1];   // (32, 64, 7)
    const float* heat = (const float*)d_in[2];   // (32, 64, 128, 128)
    float* per_obj    = (float*)d_ws;            // 2048 floats of scratch
    float* out        = (float*)d_out;           // 32 floats

    argmax_bce_kernel<<<NB * NJ, 256, 0, stream>>>(pred, gt, heat, per_obj);
    mean_wmma_kernel<<<1, 32, 0, stream>>>(per_obj, out);
}